// HPA_10539849744779
// MI455X (gfx1250) — compile-verified
//
#include <hip/hip_runtime.h>
#include <float.h>
#include <stdint.h>

// Problem constants (fixed by the reference).
#define NB    512        // B*GROUPS blocks
#define CG    8          // channels per group
#define HH    80
#define WW    80
#define HW    6400       // H*W
#define LL    160        // concat length (H + W)
#define BLK_ELEMS (CG*HW)        // 51200 floats per block tile
#define NTHREADS 1024

typedef float v2f __attribute__((ext_vector_type(2)));
typedef float v8f __attribute__((ext_vector_type(8)));

// LDS layout (floats):
//   s_gx       [51200]
//   s_cat_mean [8*160]
//   s_cat_max  [8*160]
//   s_sg_x     [8*160]   sigmoid gates, mean branch ([c][0..79]=gh, [c][80..159]=gw)
//   s_sg_y     [8*160]   sigmoid gates, max branch
//   s_red      [8*4*6]   per-(channel,wave4) partials: sumx,ssqx,sumy,ssqy,maxy,miny
//   s_ch       [57]      Ax[8],Bx[8],Ay[8],By[8],maxy1[8],coefX[8],coefY[8],wconst
#define SMEM_FLOATS (BLK_ELEMS + 4*(CG*LL) + (CG*4*6) + 57)
#define SMEM_BYTES  (SMEM_FLOATS * 4)

__device__ __forceinline__ float sigmoidf(float v) {
    // v_rcp_f32-based: ~1 ulp, avoids the IEEE div_scale/fma expansion.
    return __builtin_amdgcn_rcpf(1.0f + __expf(-v));
}

__global__ __launch_bounds__(NTHREADS, 1)
void HPA_10539849744779_kernel(const float* __restrict__ x,
                               const float* __restrict__ cw,
                               const float* __restrict__ cb,
                               const float* __restrict__ gnw,
                               const float* __restrict__ gnb,
                               float* __restrict__ out) {
    extern __shared__ float smem[];
    float* s_gx       = smem;
    float* s_cat_mean = s_gx + BLK_ELEMS;
    float* s_cat_max  = s_cat_mean + CG*LL;
    float* s_sg_x     = s_cat_max  + CG*LL;
    float* s_sg_y     = s_sg_x     + CG*LL;
    float* s_red      = s_sg_y     + CG*LL;
    float* s_ch       = s_red      + CG*4*6;

    const int t  = threadIdx.x;
    const int nb = blockIdx.x;

    // ---- Pass 0: stage the 8x80x80 tile into LDS via async HBM->LDS DMA ----
    // global_load_async_to_lds_b128 bypasses VGPRs entirely (ASYNCcnt-tracked).
    // Flat LDS pointers carry the DS offset in addr[31:0] (aperture rule).
    {
        const uint64_t gbase = (uint64_t)(size_t)(x + (size_t)nb * BLK_ELEMS);
        const uint32_t lbase = (uint32_t)(size_t)s_gx;
        for (int i = t; i < BLK_ELEMS / 4; i += NTHREADS) {
            const uint32_t boff  = (uint32_t)i * 16u;
            const uint32_t laddr = lbase + boff;
            const uint64_t gaddr = gbase + boff;
            asm volatile("global_load_async_to_lds_b128 %0, %1, off"
                         :: "v"(laddr), "v"(gaddr) : "memory");
        }
        asm volatile("s_wait_asynccnt 0x0" ::: "memory");
    }
    __syncthreads();

    // ---- Pass A: row (over w) and column (over h) mean/max pooling ----
    for (int task = t; task < 2 * CG * HH; task += NTHREADS) {
        if (task < CG * HH) {
            int c = task / HH, h = task - c * HH;
            const float* p = s_gx + c * HW + h * WW;
            float s = 0.0f, m = -FLT_MAX;
            for (int w = 0; w < WW; ++w) { float v = p[w]; s += v; m = fmaxf(m, v); }
            s_cat_mean[c * LL + h] = s * (1.0f / WW);
            s_cat_max [c * LL + h] = m;
        } else {
            int tt = task - CG * HH;
            int c = tt / WW, w = tt - c * WW;
            const float* p = s_gx + c * HW + w;
            float s = 0.0f, m = -FLT_MAX;
            for (int h = 0; h < HH; ++h) { float v = p[h * WW]; s += v; m = fmaxf(m, v); }
            s_cat_mean[c * LL + HH + w] = s * (1.0f / HH);
            s_cat_max [c * LL + HH + w] = m;
        }
    }
    __syncthreads();

    // ---- Pass M: 8x8 conv over 160 positions per branch via WMMA f32 16x16x4 ----
    // hw[o,l] = sum_c cw[o,c]*cat[c,l] + cb[o]; store sigmoid(hw) gates.
    // A (16x4): lanes 0-15 row m; VGPR0 = K0|K2 (lane halves), VGPR1 = K1|K3.
    // Rows 8..15 zero-padded. Two K=4 chunks chained through the accumulator.
    if (t < 32) {
        const int lane = t;
        const int m = lane & 15;
        const bool hi = lane >= 16;
        const int k0 = hi ? 2 : 0, k1 = hi ? 3 : 1;

        v2f a0, a1;
        a0.x = (m < CG) ? cw[m * CG + k0]     : 0.0f;
        a0.y = (m < CG) ? cw[m * CG + k1]     : 0.0f;
        a1.x = (m < CG) ? cw[m * CG + 4 + k0] : 0.0f;
        a1.y = (m < CG) ? cw[m * CG + 4 + k1] : 0.0f;

        v8f cacc;
        #pragma unroll
        for (int r = 0; r < 8; ++r) cacc[r] = hi ? 0.0f : cb[r];  // bias in rows 0..7

        #pragma unroll
        for (int br = 0; br < 2; ++br) {
            const float* cat = br ? s_cat_max : s_cat_mean;
            float* sg = br ? s_sg_y : s_sg_x;
            for (int T = 0; T < LL / 16; ++T) {
                const int col = T * 16 + m;
                v2f b0, b1;
                b0.x = cat[k0 * LL + col];
                b0.y = cat[k1 * LL + col];
                b1.x = cat[(4 + k0) * LL + col];
                b1.y = cat[(4 + k1) * LL + col];
                v8f d = __builtin_amdgcn_wmma_f32_16x16x4_f32(
                    false, a0, false, b0, (short)0, cacc, false, false);
                d = __builtin_amdgcn_wmma_f32_16x16x4_f32(
                    false, a1, false, b1, (short)0, d, false, false);
                if (!hi) {
                    #pragma unroll
                    for (int r = 0; r < 8; ++r)
                        sg[r * LL + col] = sigmoidf(d[r]);
                }
            }
        }
    }
    __syncthreads();

    // ---- Pass B: gated stats per channel (float4-vectorized LDS reads) ----
    {
        const int c  = t >> 7;        // 128 threads (4 waves) per channel
        const int s0 = t & 127;
        const float4* gp4 = (const float4*)(s_gx + c * HW);
        const float* gtx = s_sg_x + c * LL;
        const float* gty = s_sg_y + c * LL;

        float sumx = 0.f, ssqx = 0.f, sumy = 0.f, ssqy = 0.f;
        float mxy = -FLT_MAX, mny = FLT_MAX;
        for (int i = 0; i < 13; ++i) {
            int q = s0 + (i << 7);                 // float4 index, < 1600
            if (q < HW / 4) {
                int l = q << 2;
                int h = l / WW;                    // 80%4==0: no row crossing
                int w = l - h * WW;
                float4 g = gp4[q];
                float xh = gtx[h], yh = gty[h];
                const float* xw = gtx + HH + w;
                const float* yw = gty + HH + w;
                float gx0 = g.x * xh * xw[0], gy0 = g.x * yh * yw[0];
                float gx1 = g.y * xh * xw[1], gy1 = g.y * yh * yw[1];
                float gx2 = g.z * xh * xw[2], gy2 = g.z * yh * yw[2];
                float gx3 = g.w * xh * xw[3], gy3 = g.w * yh * yw[3];
                sumx += (gx0 + gx1) + (gx2 + gx3);
                ssqx += gx0*gx0 + gx1*gx1 + gx2*gx2 + gx3*gx3;
                sumy += (gy0 + gy1) + (gy2 + gy3);
                ssqy += gy0*gy0 + gy1*gy1 + gy2*gy2 + gy3*gy3;
                mxy = fmaxf(mxy, fmaxf(fmaxf(gy0, gy1), fmaxf(gy2, gy3)));
                mny = fminf(mny, fminf(fminf(gy0, gy1), fminf(gy2, gy3)));
            }
        }
        // wave32 reduction (all lanes of a wave share c)
        #pragma unroll
        for (int off = 16; off > 0; off >>= 1) {
            sumx += __shfl_xor(sumx, off, 32);
            ssqx += __shfl_xor(ssqx, off, 32);
            sumy += __shfl_xor(sumy, off, 32);
            ssqy += __shfl_xor(ssqy, off, 32);
            mxy = fmaxf(mxy, __shfl_xor(mxy, off, 32));
            mny = fminf(mny, __shfl_xor(mny, off, 32));
        }
        if ((t & 31) == 0) {
            int w4 = (t >> 5) & 3;
            float* r = s_red + (c * 4 + w4) * 6;
            r[0] = sumx; r[1] = ssqx; r[2] = sumy; r[3] = ssqy; r[4] = mxy; r[5] = mny;
        }
    }
    __syncthreads();

    // ---- Finalize per-channel affine params ----
    if (t < CG) {
        float sx = 0.f, qx = 0.f, sy = 0.f, qy = 0.f, my = -FLT_MAX, ny = FLT_MAX;
        for (int j = 0; j < 4; ++j) {
            const float* r = s_red + (t * 4 + j) * 6;
            sx += r[0]; qx += r[1]; sy += r[2]; qy += r[3];
            my = fmaxf(my, r[4]); ny = fminf(ny, r[5]);
        }
        const float inv = 1.0f / (float)HW;
        float mux = sx * inv, varx = fmaxf(qx * inv - mux * mux, 0.0f);
        float muy = sy * inv, vary = fmaxf(qy * inv - muy * muy, 0.0f);
        float gw_ = gnw[t], gb_ = gnb[t];
        float Ax = gw_ * rsqrtf(varx + 1e-5f);
        float Bx = gb_ - Ax * mux;
        float Ay = gw_ * rsqrtf(vary + 1e-5f);
        float By = gb_ - Ay * muy;
        // max over (H,W) of y1 = Ay*gated_y + By (affine; pick max or min by sign)
        float maxy1 = (Ay >= 0.0f) ? (Ay * my + By) : (Ay * ny + By);
        s_ch[t]      = Ax;
        s_ch[8 + t]  = Bx;
        s_ch[16 + t] = Ay;
        s_ch[24 + t] = By;
        s_ch[32 + t] = maxy1;
    }
    __syncthreads();

    // ---- Softmaxes + fused weight coefficients (trivial, one thread) ----
    if (t == 0) {
        // x12 = softmax(mean(x1)) = softmax(gn_b)   (mean of normalized == gn_b exactly)
        // y12 = softmax(max(y1))
        float xb[CG], yb[CG];
        float mx = -FLT_MAX, my = -FLT_MAX;
        for (int c = 0; c < CG; ++c) {
            xb[c] = gnb[c];       mx = fmaxf(mx, xb[c]);
            yb[c] = s_ch[32 + c]; my = fmaxf(my, yb[c]);
        }
        float sx = 0.f, sy = 0.f;
        for (int c = 0; c < CG; ++c) {
            xb[c] = __expf(xb[c] - mx); sx += xb[c];
            yb[c] = __expf(yb[c] - my); sy += yb[c];
        }
        float rsx = __builtin_amdgcn_rcpf(sx), rsy = __builtin_amdgcn_rcpf(sy);
        float wc = 0.f;
        for (int c = 0; c < CG; ++c) {
            float x12 = xb[c] * rsx, y12 = yb[c] * rsy;
            s_ch[40 + c] = y12 * s_ch[c];        // coefX: multiplies gated_x
            s_ch[48 + c] = x12 * s_ch[16 + c];   // coefY: multiplies gated_y
            wc += x12 * s_ch[24 + c] + y12 * s_ch[8 + c];   // x12*By + y12*Bx
        }
        s_ch[56] = wc;
    }
    __syncthreads();

    // ---- Pass C: weights + output, float4-vectorized (b128 LDS loads, b128 stores) ----
    {
        float coefX[CG], coefY[CG];
        #pragma unroll
        for (int c = 0; c < CG; ++c) { coefX[c] = s_ch[40 + c]; coefY[c] = s_ch[48 + c]; }
        const float wconst = s_ch[56];
        float* outg = out + (size_t)nb * BLK_ELEMS;

        for (int q = t; q < HW / 4; q += NTHREADS) {
            int l = q << 2;
            int h = l / WW;
            int w = l - h * WW;
            float a0 = wconst, a1 = wconst, a2 = wconst, a3 = wconst;
            float4 gv[CG];
            #pragma unroll
            for (int c = 0; c < CG; ++c) {
                float4 g = *(const float4*)(s_gx + c * HW + l);
                gv[c] = g;
                float  xh = s_sg_x[c * LL + h];
                float  yh = s_sg_y[c * LL + h];
                float4 xw = *(const float4*)(s_sg_x + c * LL + HH + w);
                float4 yw = *(const float4*)(s_sg_y + c * LL + HH + w);
                float cx = coefX[c], cy = coefY[c];
                a0 += cx * (g.x * xh * xw.x) + cy * (g.x * yh * yw.x);
                a1 += cx * (g.y * xh * xw.y) + cy * (g.y * yh * yw.y);
                a2 += cx * (g.z * xh * xw.z) + cy * (g.z * yh * yw.z);
                a3 += cx * (g.w * xh * xw.w) + cy * (g.w * yh * yw.w);
            }
            float s0f = sigmoidf(a0), s1f = sigmoidf(a1);
            float s2f = sigmoidf(a2), s3f = sigmoidf(a3);
            #pragma unroll
            for (int c = 0; c < CG; ++c) {
                float4 o = { gv[c].x * s0f, gv[c].y * s1f, gv[c].z * s2f, gv[c].w * s3f };
                *(float4*)(outg + c * HW + l) = o;
            }
        }
    }
}

extern "C" void kernel_launch(void* const* d_in, const int* in_sizes, int n_in,
                              void* d_out, int out_size, void* d_ws, size_t ws_size,
                              hipStream_t stream) {
    (void)in_sizes; (void)n_in; (void)d_ws; (void)ws_size; (void)out_size;
    const float* x   = (const float*)d_in[0];
    const float* cw  = (const float*)d_in[1];
    const float* cb  = (const float*)d_in[2];
    const float* gnw = (const float*)d_in[3];
    const float* gnb = (const float*)d_in[4];
    float* out = (float*)d_out;

    // Allow >64KB dynamic LDS (CDNA5 WGP has 320KB).
    (void)hipFuncSetAttribute((const void*)HPA_10539849744779_kernel,
                              hipFuncAttributeMaxDynamicSharedMemorySize, SMEM_BYTES);

    HPA_10539849744779_kernel<<<dim3(NB), dim3(NTHREADS), SMEM_BYTES, stream>>>(
        x, cw, cb, gnw, gnb, out);
}